// MultiResolutionHashEncoding_77773267796436
// MI455X (gfx1250) — compile-verified
//
#include <hip/hip_runtime.h>
#include <cmath>

// ---------------- problem constants (from reference) ----------------
#define TABLE_SZ    16384
#define FEAT_DIM    2
#define NUM_LEVELS  16
#define NPTS        (1 << 20)

#define THREADS         256            // 8 wave32s per block
#define PTS_PER_BLOCK   8192           // staging (128KB) amortized 4x vs gather bytes
#define PTS_PER_THREAD  (PTS_PER_BLOCK / THREADS)   // 32
#define STAGE_ITERS     (TABLE_SZ * FEAT_DIM / 4 / THREADS)  // 32 b128 per thread
#define HASH_P1         2654435761u
#define HASH_P2         805459861u
#define LDS_BYTES       (TABLE_SZ * FEAT_DIM * sizeof(float))  // 131072

// ---------------- CDNA5 async global->LDS path (guarded) ----------------
#if defined(__has_builtin)
#  if __has_builtin(__builtin_amdgcn_global_load_async_to_lds_b128)
#    define HAVE_ASYNC_LDS 1
#  endif
#  if __has_builtin(__builtin_amdgcn_s_wait_asynccnt)
#    define HAVE_WAIT_ASYNC 1
#  endif
#endif

// pointer types matching the builtin's signature: v4i addrspace(1)* / addrspace(3)*
typedef int v4i __attribute__((ext_vector_type(4)));
typedef __attribute__((address_space(1))) v4i gv4i;   // global
typedef __attribute__((address_space(3))) v4i lv4i;   // LDS

struct LevelRes { float r[NUM_LEVELS]; };

__global__ __launch_bounds__(THREADS)
void MultiResolutionHashEncoding_77773267796436_kernel(
    const float* __restrict__ x,        // [NPTS, 3]
    const float* __restrict__ tables,   // [NUM_LEVELS, TABLE_SZ, FEAT_DIM]
    float* __restrict__ out,            // [NPTS, NUM_LEVELS*FEAT_DIM]
    LevelRes lr) {
  extern __shared__ float lds_raw[];                 // 128 KB: this level's table
  float2* __restrict__ tab = (float2*)lds_raw;

  const int lvl = blockIdx.y;
  const float r = lr.r[lvl];
  const float* __restrict__ tsrc =
      tables + (size_t)lvl * (TABLE_SZ * FEAT_DIM);

  // ---- stage this level's hash table (128 KB) into LDS ----
  // Exactly STAGE_ITERS (=32) b128 transfers per thread; fully unrolled
  // straight-line issue, no data-dependent control flow around the async ops.
  {
    const float4* __restrict__ src4 = (const float4*)tsrc + threadIdx.x;
    float4* __restrict__ dst4 = (float4*)lds_raw + threadIdx.x;
#if defined(HAVE_ASYNC_LDS)
#pragma unroll
    for (int k = 0; k < STAGE_ITERS; ++k) {
      __builtin_amdgcn_global_load_async_to_lds_b128(
          (gv4i*)(src4 + k * THREADS),
          (lv4i*)(dst4 + k * THREADS),
          /*offset=*/0, /*cpol=*/0);     // 32 outstanding <= ASYNCcnt max 63
    }
#  if defined(HAVE_WAIT_ASYNC)
    __builtin_amdgcn_s_wait_asynccnt(0);
#  endif
#else
#pragma unroll
    for (int k = 0; k < STAGE_ITERS; ++k)
      dst4[k * THREADS] = src4[k * THREADS];
#endif
    __syncthreads();   // all waves' staging visible before gathers
  }

  const int base = blockIdx.x * PTS_PER_BLOCK + threadIdx.x;

#pragma unroll 4
  for (int i = 0; i < PTS_PER_THREAD; ++i) {
    const int p = base + i * THREADS;
    // hint the next iteration's coordinates toward the cache
    if (i + 1 < PTS_PER_THREAD)
      __builtin_prefetch(&x[3 * (size_t)(p + THREADS)], 0, 0);

    const float px = x[3 * (size_t)p + 0];
    const float py = x[3 * (size_t)p + 1];
    const float pz = x[3 * (size_t)p + 2];

    // interp weights from the UNSCALED position (faithful to reference)
    const float wx = px - floorf(px);
    const float wy = py - floorf(py);
    const float wz = pz - floorf(pz);
    const float omx = 1.0f - wx, omy = 1.0f - wy, omz = 1.0f - wz;

    const float sx = r * px, sy = r * py, sz = r * pz;
    const unsigned ulx = (unsigned)(int)floorf(sx);
    const unsigned uly = (unsigned)(int)floorf(sy);
    const unsigned ulz = (unsigned)(int)floorf(sz);
    const unsigned uhx = (unsigned)(int)ceilf(sx);
    const unsigned uhy = (unsigned)(int)ceilf(sy);
    const unsigned uhz = (unsigned)(int)ceilf(sz);

    float f0 = 0.0f, f1 = 0.0f;

    // corner order v0..v7 per reference OFFSETS (z-ascending, ccw)
#define CORNER(OX, OY, OZ)                                                  \
    {                                                                       \
      const unsigned cx = (OX) ? uhx : ulx;                                 \
      const unsigned cy = (OY) ? uhy : uly;                                 \
      const unsigned cz = (OZ) ? uhz : ulz;                                 \
      const unsigned h =                                                    \
          (cx ^ (HASH_P1 * cy) ^ (HASH_P2 * cz)) & (TABLE_SZ - 1u);         \
      const float2 f = tab[h];                        /* ds_load_b64 */     \
      const float cw = ((OX) ? wx : omx) * ((OY) ? wy : omy) *              \
                       ((OZ) ? wz : omz);                                   \
      f0 = fmaf(cw, f.x, f0);                                               \
      f1 = fmaf(cw, f.y, f1);                                               \
    }
    CORNER(0, 0, 0) CORNER(1, 0, 0) CORNER(1, 1, 0) CORNER(0, 1, 0)
    CORNER(0, 0, 1) CORNER(1, 0, 1) CORNER(1, 1, 1) CORNER(0, 1, 1)
#undef CORNER

    // out[p, lvl*2 .. lvl*2+1] — 8B store; lines combine in L2 (output fits)
    float2* o = (float2*)(out + (size_t)p * (NUM_LEVELS * FEAT_DIM) +
                          lvl * FEAT_DIM);
    *o = float2{f0, f1};
  }
}

extern "C" void kernel_launch(void* const* d_in, const int* in_sizes, int n_in,
                              void* d_out, int out_size, void* d_ws, size_t ws_size,
                              hipStream_t stream) {
  const float* x      = (const float*)d_in[0];   // [B,3] float32
  const float* tables = (const float*)d_in[1];   // [16,16384,2] float32
  float* out          = (float*)d_out;           // [B,32] float32
  (void)in_sizes; (void)n_in; (void)out_size; (void)d_ws; (void)ws_size;

  // res[l] = floor(MIN_RES * b^l), b = exp((ln 512 - ln 16)/15), in float32
  LevelRes lr;
  const float bgrow = expf((logf(512.0f) - logf(16.0f)) / 15.0f);
  for (int l = 0; l < NUM_LEVELS; ++l)
    lr.r[l] = floorf(16.0f * powf(bgrow, (float)l));

  dim3 grid(NPTS / PTS_PER_BLOCK, NUM_LEVELS);   // (128, 16)
  dim3 block(THREADS);
  MultiResolutionHashEncoding_77773267796436_kernel
      <<<grid, block, LDS_BYTES, stream>>>(x, tables, out, lr);
}